// Net_18408229830719
// MI455X (gfx1250) — compile-verified
//
#include <hip/hip_runtime.h>

typedef float v2f __attribute__((ext_vector_type(2)));
typedef float v4f __attribute__((ext_vector_type(4)));
typedef float v8f __attribute__((ext_vector_type(8)));
typedef unsigned int u32x4 __attribute__((ext_vector_type(4)));
typedef int i32x4 __attribute__((ext_vector_type(4)));
typedef int i32x8 __attribute__((ext_vector_type(8)));

#define Bsz 8192
#define Tn  256
#define Dn  32
#define Un  32
#define On  16

#define WAVES   4
#define ATS     1026              // padded LDS row stride (dwords): even (b64 align), 1026%64==2 spreads banks
#define AT_DW   (32 * ATS)        // A^T: 32 columns x 1024 k (padded, interleaved)
#define HROW_S  36                // padded h-tile row stride (dwords): 16B aligned, conflict-free b128 reads
#define HB_DW   (16 * HROW_S)
#define XB0_DW  (AT_DW + WAVES * HB_DW)   // x staging buffers start here
#define XW_DW   8192              // per-wave x staging: 2 bufs x 8 steps x 16 rows x 32 f32 = 32KB
#define SMEM_DW (XB0_DW + WAVES * XW_DW)  // ~265 KB total, fits 320 KB WGP LDS

// ---- TDM: iterated 2D tile load, 16 rows x 32 f32, row stride T*D, 8 timesteps
// per issue (global += 32 elems/iter, LDS += 512 elems/iter). D# per ISA 8.3-8.6.
__device__ __forceinline__ void tdm_load_x(const float* gptr, unsigned ldsByteOff) {
  unsigned long long ga = (unsigned long long)(uintptr_t)gptr;
  u32x4 g0 = { 1u,                                   // count=1, user mode
               ldsByteOff,                           // lds_addr (bytes)
               (unsigned)ga,                         // global_addr[31:0]
               (unsigned)((ga >> 32) & 0x1FFFFFFu) | (2u << 30) }; // global_addr[56:32] | type=2
  i32x8 g1 = { 0x000A0000,   // wg_mask=0 | data_size=4B (2<<16) | iterate_enable (1<<19)
               0x20000000,   // tensor_dim0 = 8192 -> bits[63:48]
               0x20000000,   // tensor_dim1 = 8192 -> bits[95:80]
               0x00200000,   // tile_dim0 = 32 -> bits[127:112]
               16,           // tile_dim1 = 16 rows
               Tn * Dn,      // tensor_dim0_stride = 8192 (low 32 of 48)
               0, 0 };
  i32x4 g2 = { 1,            // tensor_dim2 (unused)
               512,          // lds_addr_increment per iter (elements) = 2KB
               Dn,           // global_addr_increment per iter (elements) = 128B
               7 << 16 };    // iterate_count = 7 -> 8 iterations
  i32x4 g3 = { 0, 0, 0, 0 };
#if __clang_major__ >= 23
  i32x8 g4 = { 0, 0, 0, 0, 0, 0, 0, 0 };
  __builtin_amdgcn_tensor_load_to_lds(g0, g1, g2, g3, g4, 0);
#else
  __builtin_amdgcn_tensor_load_to_lds(g0, g1, g2, g3, 0);
#endif
}

__global__ __launch_bounds__(WAVES * 32, 1)
void krec_kernel(const float* __restrict__ x, const float* __restrict__ A,
                 const float* __restrict__ alpha, const float* __restrict__ Om,
                 float* __restrict__ out) {
  extern __shared__ float smem[];
  const int tid  = threadIdx.x;
  const int lane = tid & 31;
  const int wave = tid >> 5;
  const int row  = lane & 15;     // WMMA row / column index within 16
  const int hi   = lane >> 4;     // half-wave selector (K phase)

  // ---- Stage A^T into LDS with INTERLEAVED column rows (perm(n)=(n&15)*2+(n>>4)):
  // a lane's two N-tile columns are adjacent LDS rows -> all B-fragment loads
  // use 16-bit immediate DS offsets from one base, merged as ds_load_2addr_b64.
  for (int i = tid; i < 1024 * 32; i += WAVES * 32) {
    int k = i >> 5, n = i & 31;
    smem[((n & 15) * 2 + (n >> 4)) * ATS + k] = A[i];
  }
  __syncthreads();

  float* hb = smem + AT_DW + wave * HB_DW;             // per-wave private h tile
  const float* colbase = smem + (2 * row) * ATS + 2 * hi;

  float hrow[32];
#pragma unroll
  for (int u = 0; u < 32; ++u) hrow[u] = alpha[u];     // h0 = alpha (broadcast)

  const int b0 = (blockIdx.x * WAVES + wave) * 16;

  // wave-uniform (SGPR) TDM parameters
  const int b0s = __builtin_amdgcn_readfirstlane(b0);
  const unsigned xwByte = (unsigned)(XB0_DW + __builtin_amdgcn_readfirstlane(wave) * XW_DW) * 4u;
  const float* gx = x + (size_t)b0s * (Tn * Dn);

  // per-lane LDS read base into the x staging area
  const float* xlsw = smem + XB0_DW + wave * XW_DW + row * Dn + 2 * hi;

  tdm_load_x(gx, xwByte);                              // group 0 -> buffer 0

#pragma unroll 1
  for (int g = 0; g < Tn / 8; ++g) {                   // 32 groups of 8 timesteps
    if (g < Tn / 8 - 1) {                              // prefetch next group -> other buffer
      tdm_load_x(gx + (size_t)(g + 1) * 8 * Dn, xwByte + (unsigned)(((g + 1) & 1) * 16384));
      __builtin_amdgcn_s_wait_tensorcnt(1);            // older (current) group complete
    } else {
      __builtin_amdgcn_s_wait_tensorcnt(0);
    }
    const float* xg = xlsw + (g & 1) * (XW_DW / 2);

#pragma unroll 1
    for (int s = 0; s < 8; ++s) {
      const float* xp = xg + s * 512;                  // this step's 2KB x tile
      v2f xv[8];
#pragma unroll
      for (int c = 0; c < 8; ++c) xv[c] = *(const v2f*)(xp + 4 * c);

      v8f accA0 = {}, accA1 = {}, accB0 = {}, accB1 = {};
#pragma unroll
      for (int u = 0; u < 32; ++u) {
        const float hu = hrow[u];
#pragma unroll
        for (int c = 0; c < 8; ++c) {
          v2f af = { hu * xv[c].x, hu * xv[c].y };              // z[row, u*32+4c+2hi..+1]
          v2f bf0 = *(const v2f*)(colbase + u * 32 + 4 * c);     // A[k..k+1][col]
          v2f bf1 = *(const v2f*)(colbase + ATS + u * 32 + 4 * c); // A[k..k+1][col+16]
          if (c & 1) {
            accA1 = __builtin_amdgcn_wmma_f32_16x16x4_f32(false, af, false, bf0, (short)0, accA1, false, false);
            accB1 = __builtin_amdgcn_wmma_f32_16x16x4_f32(false, af, false, bf1, (short)0, accB1, false, false);
          } else {
            accA0 = __builtin_amdgcn_wmma_f32_16x16x4_f32(false, af, false, bf0, (short)0, accA0, false, false);
            accB0 = __builtin_amdgcn_wmma_f32_16x16x4_f32(false, af, false, bf1, (short)0, accB0, false, false);
          }
        }
      }
      v8f hA = accA0 + accA1;   // h_new cols 0..15
      v8f hB = accB0 + accB1;   // h_new cols 16..31

      // C-frag -> LDS h tile (VGPR r: lanes0-15 M=r, lanes16-31 M=8+r, N=lane&15)
#pragma unroll
      for (int r = 0; r < 8; ++r) {
        hb[(r + 8 * hi) * HROW_S + row]      = hA[r];
        hb[(r + 8 * hi) * HROW_S + row + 16] = hB[r];
      }
      // re-read per-row h for next step's A fragments (same-wave DS ops in order)
#pragma unroll
      for (int j = 0; j < 8; ++j) {
        v4f h4 = *(const v4f*)(hb + row * HROW_S + 4 * j);
        hrow[4 * j + 0] = h4.x; hrow[4 * j + 1] = h4.y;
        hrow[4 * j + 2] = h4.z; hrow[4 * j + 3] = h4.w;
      }
    }
  }

  // ---- epilogue: out = h @ Omega  (16x32 @ 32x16, single C tile)
  v8f oacc = {};
#pragma unroll
  for (int c = 0; c < 8; ++c) {
    const int k0 = 4 * c;
    v2f af = { hi ? hrow[k0 + 2] : hrow[k0 + 0],
               hi ? hrow[k0 + 3] : hrow[k0 + 1] };
    const int kk = k0 + 2 * hi;
    v2f bf = { Om[kk * On + row], Om[(kk + 1) * On + row] };
    oacc = __builtin_amdgcn_wmma_f32_16x16x4_f32(false, af, false, bf, (short)0, oacc, false, false);
  }
#pragma unroll
  for (int r = 0; r < 8; ++r)
    out[(size_t)(b0 + r + 8 * hi) * On + row] = oacc[r];
}

extern "C" void kernel_launch(void* const* d_in, const int* in_sizes, int n_in,
                              void* d_out, int out_size, void* d_ws, size_t ws_size,
                              hipStream_t stream) {
  const float* x     = (const float*)d_in[0];   // (B, T, D) fp32
  const float* A     = (const float*)d_in[1];   // (U*D, U) fp32
  const float* alpha = (const float*)d_in[2];   // (1, U)  fp32
  const float* Om    = (const float*)d_in[3];   // (U, O)  fp32
  float* out = (float*)d_out;                   // (B, O)  fp32

  dim3 grid(Bsz / (WAVES * 16));   // 128 blocks, 4 waves each -> 512 batch tiles of 16 rows
  dim3 block(WAVES * 32);
  size_t smem = (size_t)SMEM_DW * sizeof(float);   // ~265 KB of the 320 KB WGP LDS
  krec_kernel<<<grid, block, smem, stream>>>(x, A, alpha, Om, out);
}